// MyFunctionBlockFull_43276090474799
// MI455X (gfx1250) — compile-verified
//
#include <hip/hip_runtime.h>
#include <hip/hip_bf16.h>

typedef float v2f __attribute__((ext_vector_type(2)));
typedef float v4f __attribute__((ext_vector_type(4)));
typedef float v8f __attribute__((ext_vector_type(8)));

// log2|v| via hardware v_log_f32
__device__ __forceinline__ float lg2a(float v) {
    return __builtin_amdgcn_logf(__builtin_fabsf(v));
}

// One wave handles a 16-row tile.
//   lane = r + 16*half ; half=0 -> row r elements 0..7 (K-slots 0,1)
//                        half=1 -> row r elements 8..15 (K-slots 2,3)
// A (16x4 f32) = weights: row 0 = W_sum slice, row 1 = W_prod slice, rest 0.
// B (4x16 f32) = features: column n = row n's transformed values.
// D[0][n] = sum-dot(row n), D[1][n] = log2-prod-dot(row n).
__global__ __launch_bounds__(256)
void fused_feat_dot_kernel(const float* __restrict__ x,
                           const float* __restrict__ Wsum,
                           const float* __restrict__ Wprod,
                           float* __restrict__ out,
                           int ntiles)
{
    constexpr float LOG2E  = 1.44269504088896340736f;
    constexpr float LN2    = 0.69314718055994530942f;
    constexpr float INV2PI = 0.15915494309189533577f;  // v_sin/v_cos take turns

    const int lane = threadIdx.x & 31;
    const int wave = threadIdx.x >> 5;
    const int tile = blockIdx.x * (blockDim.x >> 5) + wave;
    if (tile >= ntiles) return;               // wave-uniform: EXEC stays all-ones

    const int  r    = lane & 15;              // row within tile == output column N
    const int  half = lane >> 4;              // which 8-element slice of the row
    const int  jlo  = half << 3;
    const long row0 = (long)tile << 4;

    // --- load this lane's 8 x-values (two b128, fully coalesced per wave) ---
    const float* xp = x + (row0 + r) * 16 + jlo;
    v4f xa = *(const v4f*)xp;
    v4f xb = *(const v4f*)(xp + 4);
    float xv[8] = {xa.x, xa.y, xa.z, xa.w, xb.x, xb.y, xb.z, xb.w};

    // weight array for this lane's A-row class (lanes r>=2 load valid-but-masked data)
    const float* wbase = (lane & 1) ? Wprod : Wsum;
    const float  selS  = (r == 0) ? 1.0f : 0.0f;   // A row 0 (sum path)
    const float  selP  = (r == 1) ? 1.0f : 0.0f;   // A row 1 (prod path)

    // shared transcendental precomputes: log2|x| and e^x (reused by 5 blocks)
    float lx[8], ex[8];
#pragma unroll
    for (int i = 0; i < 8; ++i) {
        lx[i] = lg2a(xv[i]);                                   // v_log_f32
        ex[i] = __builtin_amdgcn_exp2f(xv[i] * LOG2E);         // v_exp_f32
    }

    v8f accS = {0,0,0,0,0,0,0,0};   // feat-chain accumulator
    v8f accP = {0,0,0,0,0,0,0,0};   // log-feat-chain accumulator

#pragma unroll
    for (int b = 0; b < 7; ++b) {
        // per-block weight slice: 8 contiguous floats (L0-resident, 448B arrays)
        const float* wp = wbase + b * 16 + jlo;
        v4f w0v = *(const v4f*)wp;
        v4f w1v = *(const v4f*)(wp + 4);
        float w[8] = {w0v.x, w0v.y, w0v.z, w0v.w, w1v.x, w1v.y, w1v.z, w1v.w};

        // features F and their log2 magnitudes LF for this block
        float F[8], LF[8];
#pragma unroll
        for (int i = 0; i < 8; ++i) {
            const float xi = xv[i];
            if (b == 0)      { F[i] = xi;                LF[i] = lx[i]; }
            else if (b == 1) { F[i] = __builtin_amdgcn_sinf(xi * INV2PI); LF[i] = lg2a(F[i]); }
            else if (b == 2) { F[i] = __builtin_amdgcn_cosf(xi * INV2PI); LF[i] = lg2a(F[i]); }
            else if (b == 3) { const float e2 = ex[i] * ex[i];            // tanh = (e2-1)/(e2+1)
                               F[i]  = (e2 - 1.0f) * __builtin_amdgcn_rcpf(e2 + 1.0f);
                               LF[i] = lg2a(F[i]); }
            else if (b == 4) { F[i] = ex[i];             LF[i] = xi * LOG2E; } // log2(e^x)=x*log2e
            else if (b == 5) { F[i] = lx[i] * LN2;       LF[i] = lg2a(F[i]); } // ln|x|
            else             { const float rr = __builtin_amdgcn_rcpf(1.0f + ex[i]);
                               F[i]  = ex[i] * rr;                         // sigmoid
                               LF[i] = lg2a(F[i]); }
        }

        // 4 K-slices of 4 per block, two independent accumulation chains
#pragma unroll
        for (int s = 0; s < 4; ++s) {
            v2f aS; aS.x = selS * w[2*s]; aS.y = selS * w[2*s + 1];
            v2f bS; bS.x = F[2*s];        bS.y = F[2*s + 1];
            accS = __builtin_amdgcn_wmma_f32_16x16x4_f32(
                false, aS, false, bS, (short)0, accS, false, false);

            v2f aP; aP.x = selP * w[2*s]; aP.y = selP * w[2*s + 1];
            v2f bP; bP.x = LF[2*s];       bP.y = LF[2*s + 1];
            accP = __builtin_amdgcn_wmma_f32_16x16x4_f32(
                false, aP, false, bP, (short)0, accP, false, false);
        }
    }

    // D readout: VGPR0 lanes 0-15 = (M=0, N=lane) = sum ; VGPR1 = (M=1) = log2-prod
    if (lane < 16) {
        const float s = accS[0] + accP[0];                       // accP row0 == 0
        const float p = __builtin_amdgcn_exp2f(accS[1] + accP[1]);
        float2 o; o.x = s; o.y = p;
        *(float2*)(out + (row0 + lane) * 2) = o;                 // coalesced 8B store
    }
}

extern "C" void kernel_launch(void* const* d_in, const int* in_sizes, int n_in,
                              void* d_out, int out_size, void* d_ws, size_t ws_size,
                              hipStream_t stream) {
    const float* x     = (const float*)d_in[0];   // (B,16) f32
    const float* Wsum  = (const float*)d_in[1];   // (112,1) f32
    const float* Wprod = (const float*)d_in[2];   // (112,1) f32
    float* out = (float*)d_out;                   // (B,2) f32

    const int rows   = in_sizes[0] / 16;          // B
    const int ntiles = rows / 16;                 // 16 rows per wave (B=1M -> 65536)
    const int wavesPerBlock = 8;                  // 256 threads = 8 wave32
    const int grid = (ntiles + wavesPerBlock - 1) / wavesPerBlock;

    fused_feat_dot_kernel<<<grid, 256, 0, stream>>>(x, Wsum, Wprod, out, ntiles);
}